// Layers_RNN_60902636257619
// MI455X (gfx1250) — compile-verified
//
#include <hip/hip_runtime.h>
#include <cstdint>

// Problem dimensions (compile-time, from the reference).
#define L_DIM 512
#define B_DIM 64
#define C_DIM 512
#define H_DIM 1024

typedef __attribute__((ext_vector_type(2))) float v2f;
typedef __attribute__((ext_vector_type(8))) float v8f;
typedef __attribute__((ext_vector_type(4))) unsigned int u32x4;
typedef __attribute__((ext_vector_type(8))) int i32x8;
typedef __attribute__((ext_vector_type(4))) int i32x4;

// fp32 WMMA helper: D = A(16x4) x B(4x16) + C, f32 accumulate.
__device__ inline v8f wmma4(v2f a, v2f b, v8f c) {
    return __builtin_amdgcn_wmma_f32_16x16x4_f32(
        /*neg_a=*/false, a, /*neg_b=*/false, b,
        /*c_mod=*/(short)0, c, /*reuse_a=*/false, /*reuse_b=*/false);
}

// Low 32 bits of a flat shared-memory address == LDS byte offset (ISA 10.2:
// LDS aperture keeps the offset in addr[31:0]).
__device__ inline unsigned lds_offset(const void* p) {
    return (unsigned)(uintptr_t)p;
}

// LDS row pitch for staged tiles: 64 data DWORDs + 4 pad DWORDs inserted by
// the TDM (bank-conflict avoidance: pitch%64 == 4 -> the 16 fragment lanes hit
// 16 distinct banks; a full 32-lane b64 read covers all 64 banks once).
#define LDS_PITCH 68

// -----------------------------------------------------------------------------
// Issue a TDM 2D tile load: rows x 64 fp32 tile, row stride `stride` elements,
// packed into LDS at lds_off with 4 pad DWORDs after every 64 stored DWORDs
// (i.e., per row, since tile_dim0 == 64).  Descriptor layout per ISA 8.3/8.4:
//   g0: [1:0]=count=1, [63:32]=lds_addr, [120:64]=global_addr, [127:126]=type=2
//   g1: [17:16]=data_size=2 (4B), [20]=pad_enable, [24:22]=pad_interval(5 = 64
//       DWORDs), [31:25]=pad_amount(3 = 4 DWORDs), [79:48]=tensor_dim0=64,
//       [111:80]=tensor_dim1=rows, [127:112]=tile_dim0=64, [143:128]=tile_dim1
//       =rows, [207:160]=tensor_dim0_stride
// tensor dims == tile dims -> no OOB clipping.  Tracked by TENSORcnt.
// 6-arg builtin form: (u32x4 g0, i32x8 g1, i32x4 g2, i32x4 g3, i32x8, i32 cpol)
// -----------------------------------------------------------------------------
__device__ inline void tdm_load_2d(const float* gsrc, unsigned lds_off,
                                   int rows, int stride) {
    constexpr int cols = 64;
    unsigned long long ga = (unsigned long long)(uintptr_t)gsrc;
    u32x4 g0;
    g0.x = 1u;                                             // count=1, user mode
    g0.y = lds_off;                                        // LDS byte address
    g0.z = (unsigned)ga;                                   // global addr [31:0]
    g0.w = (unsigned)((ga >> 32) & 0x01FFFFFFull) | (2u << 30); // addr[56:32], type=2
    i32x8 g1;
    g1[0] = (int)((2u << 16)        // data_size = 4 bytes
                | (1u << 20)        // pad_enable
                | (5u << 22)        // pad_interval: every 64 DWORDs
                | (3u << 25));      // pad_amount:   4 DWORDs
    g1[1] = (int)(((unsigned)cols & 0xFFFFu) << 16);       // tensor_dim0 lo16
    g1[2] = (int)((((unsigned)cols >> 16) & 0xFFFFu) |
                  (((unsigned)rows & 0xFFFFu) << 16));     // tdim0 hi / tdim1 lo
    g1[3] = (int)((((unsigned)rows >> 16) & 0xFFFFu) |
                  (((unsigned)cols) << 16));               // tdim1 hi / tile_dim0
    g1[4] = (int)((unsigned)rows & 0xFFFFu);               // tile_dim1 (tile_dim2=0)
    g1[5] = (int)stride;                                   // tensor_dim0_stride lo32
    g1[6] = 0;
    g1[7] = 0;
    i32x4 z4 = {0, 0, 0, 0};
    i32x8 z8 = {0, 0, 0, 0, 0, 0, 0, 0};
    __builtin_amdgcn_tensor_load_to_lds(g0, g1, z4, z4, z8, 0);
}

// -----------------------------------------------------------------------------
// Phase 1: xp = x @ W^T + bias   (M=L*B=32768, N=H=1024, K=C=512; lda==ldw==K)
// Block = 8 waves covering a 32(M) x 64(N) macro-tile, one 16x16 WMMA tile per
// wave.  K is staged in 64-wide chunks via TDM into double-buffered LDS
// (A: 32x68, W: 64x68 with pad), synchronized with s_wait_tensorcnt + barrier;
// WMMA fragments come from LDS (conflict-free ds_load_b64).
// -----------------------------------------------------------------------------
__global__ __launch_bounds__(256) void wmma_gemm_tdm(
    const float* __restrict__ A,      // [M, K]
    const float* __restrict__ W,      // [N, K]
    const float* __restrict__ bias,   // [N]
    float* __restrict__ D,            // [M, N]
    int K, int N, int nbn)            // nbn = N/64
{
    constexpr int KC = 64;
    __shared__ float Abuf[2][32 * LDS_PITCH];   // 2 x 8.5 KB
    __shared__ float Bbuf[2][64 * LDS_PITCH];   // 2 x 17 KB

    const int lane = threadIdx.x & 31;
    const int wave = threadIdx.x >> 5;
    const int half = lane >> 4;
    const int r    = lane & 15;
    const int wm   = wave >> 2;          // 0..1
    const int wn   = wave & 3;           // 0..3
    const int bm   = blockIdx.x / nbn;
    const int bn   = blockIdx.x % nbn;

    const float* Ablk = A + (size_t)bm * 32 * K;
    const float* Wblk = W + (size_t)bn * 64 * K;

    const int nk = K / KC;
    if (wave == 0) {
        tdm_load_2d(Ablk, lds_offset(&Abuf[0][0]), 32, K);
        tdm_load_2d(Wblk, lds_offset(&Bbuf[0][0]), 64, K);
    }

    v8f c = {};
    for (int kc = 0; kc < nk; ++kc) {
        const int buf = kc & 1;
        if (wave == 0) {
            if (kc + 1 < nk) {
                tdm_load_2d(Ablk + (kc + 1) * KC, lds_offset(&Abuf[buf ^ 1][0]), 32, K);
                tdm_load_2d(Wblk + (kc + 1) * KC, lds_offset(&Bbuf[buf ^ 1][0]), 64, K);
                __builtin_amdgcn_s_wait_tensorcnt((short)2);  // chunk kc complete
            } else {
                __builtin_amdgcn_s_wait_tensorcnt((short)0);
            }
        }
        __syncthreads();                  // LDS chunk kc visible to all waves
        const float* Al = &Abuf[buf][(wm * 16 + r) * LDS_PITCH + 2 * half];
        const float* Bl = &Bbuf[buf][(wn * 16 + r) * LDS_PITCH + 2 * half];
#pragma unroll 4
        for (int k = 0; k < KC; k += 4) {
            v2f a = *(const v2f*)(Al + k);
            v2f b = *(const v2f*)(Bl + k);
            c = wmma4(a, b, c);
        }
        __syncthreads();                  // done reading before buf is reused
    }

    const int col = bn * 64 + wn * 16 + r;
    const float bv = bias[col];
#pragma unroll
    for (int j = 0; j < 8; ++j) {
        const int row = bm * 32 + wm * 16 + j + 8 * half;
        D[(size_t)row * N + col] = c[j] + bv;
    }
}

// -----------------------------------------------------------------------------
// Fused pipelined recurrent step.  At launch p (0..512):
//   blocks [0,32):  layer-0 step t=p:    out0[t] = tanh(xp0[t] + h0 @ W_hh0^T)
//   blocks [32,96): layer-1 step s=p-1:  out1[s] = tanh(b1 + out0[s] @ W_ih1^T
//                                                        + h1 @ W_hh1^T)
// Layer-1 tiles use two partner waves (one per K=1024 reduction source),
// combined through LDS.  Everything layer-1 needs was produced at launch p-1.
// Direct global loads (L2-resident working set): no barriers on the latency-
// critical path.
// -----------------------------------------------------------------------------
__global__ __launch_bounds__(256) void rnn_step_fused(
    const float* __restrict__ state,   // [2, B, H]
    const float* __restrict__ xp0,     // [L, B, H]
    float* __restrict__ out0,          // [L, B, H]
    float* __restrict__ out1,          // [L, B, H] (== d_out)
    const float* __restrict__ W_hh0,   // [H, H]
    const float* __restrict__ W_ih1,   // [H, H]
    const float* __restrict__ W_hh1,   // [H, H]
    const float* __restrict__ b1,      // [H]
    int p)
{
    __shared__ float partial[4 * 256];             // 4 tiles x 16x16
    const int lane = threadIdx.x & 31;
    const int wave = threadIdx.x >> 5;
    const int half = lane >> 4;
    const int r    = lane & 15;
    const size_t SE = (size_t)B_DIM * H_DIM;

    if (blockIdx.x < 32) {
        // ---------------- layer 0, step t = p ----------------
        if (p >= L_DIM) return;
        const int t  = p;
        const int gw = blockIdx.x * 8 + wave;      // 0..255 tiles (4 x 64)
        const int tn = gw & 63, tm = gw >> 6;
        const float* hp = (t == 0) ? state : out0 + (size_t)(t - 1) * SE;
        const float* Arow = hp    + (size_t)(tm * 16 + r) * H_DIM + 2 * half;
        const float* Wrow = W_hh0 + (size_t)(tn * 16 + r) * H_DIM + 2 * half;
        v8f c = {};
#pragma unroll 4
        for (int k = 0; k < H_DIM; k += 4) {
            v2f a = *(const v2f*)(Arow + k);
            v2f b = *(const v2f*)(Wrow + k);
            c = wmma4(a, b, c);
        }
        const float* xt = xp0  + (size_t)t * SE;
        float*       Dt = out0 + (size_t)t * SE;
        const int col = tn * 16 + r;
#pragma unroll
        for (int j = 0; j < 8; ++j) {
            const int row = tm * 16 + j + 8 * half;
            const size_t o = (size_t)row * H_DIM + col;
            Dt[o] = tanhf(c[j] + xt[o]);
        }
    } else {
        // ---------------- layer 1, step s = p-1 ----------------
        if (p < 1) return;
        const int s  = p - 1;
        const int gw = (blockIdx.x - 32) * 8 + wave;  // 0..511 (2 waves/tile)
        const int tile  = gw >> 1;
        const int which = gw & 1;                     // 0: x-proj, 1: recurrent
        const int tn = tile & 63, tm = tile >> 6;
        const float* Asrc;
        const float* Wm;
        if (which == 0) { Asrc = out0 + (size_t)s * SE; Wm = W_ih1; }
        else {
            Asrc = (s == 0) ? state + SE : out1 + (size_t)(s - 1) * SE;
            Wm = W_hh1;
        }
        const float* Arow = Asrc + (size_t)(tm * 16 + r) * H_DIM + 2 * half;
        const float* Wrow = Wm   + (size_t)(tn * 16 + r) * H_DIM + 2 * half;
        v8f c = {};
#pragma unroll 4
        for (int k = 0; k < H_DIM; k += 4) {
            v2f a = *(const v2f*)(Arow + k);
            v2f b = *(const v2f*)(Wrow + k);
            c = wmma4(a, b, c);
        }
        const int tl = wave >> 1;                     // tile slot in block: 0..3
        if (which == 1) {
#pragma unroll
            for (int j = 0; j < 8; ++j)
                partial[tl * 256 + (j + 8 * half) * 16 + r] = c[j];
        }
        __syncthreads();
        if (which == 0) {
            const int col = tn * 16 + r;
            const float bv = b1[col];
            float* Dt = out1 + (size_t)s * SE;
#pragma unroll
            for (int j = 0; j < 8; ++j) {
                const int row = tm * 16 + j + 8 * half;
                const float v = c[j] + bv + partial[tl * 256 + (j + 8 * half) * 16 + r];
                Dt[(size_t)row * H_DIM + col] = tanhf(v);
            }
        }
    }
}

// -----------------------------------------------------------------------------
// Orchestration:
//   1 launch : xp0 = x @ W_ih0^T + b0           (TDM-staged WMMA GEMM)
//   513 launches: fused pipelined steps (layer0 step p || layer1 step p-1)
//   2 D2D copies: final states = last timestep of each layer
// Workspace: xp0 (128 MB) + out0 (128 MB) = 256 MB.
// -----------------------------------------------------------------------------
extern "C" void kernel_launch(void* const* d_in, const int* in_sizes, int n_in,
                              void* d_out, int out_size, void* d_ws, size_t ws_size,
                              hipStream_t stream) {
    (void)in_sizes; (void)n_in; (void)out_size; (void)ws_size;

    const float* x     = (const float*)d_in[0];
    const float* state = (const float*)d_in[1];
    const float* W_ih0 = (const float*)d_in[2];
    const float* W_hh0 = (const float*)d_in[3];
    const float* b0    = (const float*)d_in[4];
    const float* W_ih1 = (const float*)d_in[5];
    const float* W_hh1 = (const float*)d_in[6];
    const float* b1    = (const float*)d_in[7];

    float* out1 = (float*)d_out;                          // [L, B, H]
    float* xp   = (float*)d_ws;                           // [L*B, H]
    float* out0 = xp + (size_t)L_DIM * B_DIM * H_DIM;     // [L, B, H]

    const size_t step_elems = (size_t)B_DIM * H_DIM;
    const int nbn = H_DIM / 64;                           // 16
    const int blocks_big = (L_DIM * B_DIM / 32) * nbn;    // 16384

    // Phase 1: layer-0 input projection.
    wmma_gemm_tdm<<<blocks_big, 256, 0, stream>>>(
        x, W_ih0, b0, xp, C_DIM, H_DIM, nbn);

    // Phase 2: pipelined two-layer recurrence (513 launches instead of 1024).
    for (int p = 0; p <= L_DIM; ++p) {
        rnn_step_fused<<<96, 256, 0, stream>>>(
            state, xp, out0, out1, W_hh0, W_ih1, W_hh1, b1, p);
    }

    // Final states appended after out1.
    float* fs = out1 + (size_t)L_DIM * step_elems;
    (void)hipMemcpyAsync(fs, out0 + (size_t)(L_DIM - 1) * step_elems,
                         step_elems * sizeof(float), hipMemcpyDeviceToDevice, stream);
    (void)hipMemcpyAsync(fs + step_elems, out1 + (size_t)(L_DIM - 1) * step_elems,
                         step_elems * sizeof(float), hipMemcpyDeviceToDevice, stream);
}